// SIFTExtractor_39058432589919
// MI455X (gfx1250) — compile-verified
//
#include <hip/hip_runtime.h>
#include <hip/hip_bf16.h>
#include <math.h>

// ---------------------------------------------------------------------------
// SIFT/HOG extractor for MI455X (gfx1250, wave32)
//   x:     [bs, 3, 224, 224] f32
//   gauss: [1, 64] f32
//   out:   [bs, 28*28*8] f32, L2-normalized per image
// ---------------------------------------------------------------------------

#define IMG    224
#define CS     8
#define HC     28            // IMG / CS
#define NBINS  8
#define CELLS_PER_STRIP 28   // one cell-row per workgroup
#define HIST_PER_IMG (HC * HC * NBINS)   // 6272
#define TWO_PI_F 6.283185307179586f
#define INV_BINW 1.2732395447351628f     // NBINS / (2*pi) = 4/pi

typedef __attribute__((ext_vector_type(2))) float v2f;
typedef __attribute__((ext_vector_type(8))) float v8f;

// ---------------------------------------------------------------------------
// Kernel 1: fused gray -> sobel -> mag/angle -> gaussian-weighted LDS histogram
// grid.x = bs * 28 (image, cell-row strip); block = 256 threads (8 waves)
// ---------------------------------------------------------------------------
__global__ __launch_bounds__(256) void hog_strip_kernel(
    const float* __restrict__ x,
    const float* __restrict__ gauss,
    float* __restrict__ out)
{
    // LDS: gray halo tile 10 rows x 226 cols (stride 228), per-strip histogram,
    // gaussian weights.
    __shared__ float sgray[10 * 228];
    __shared__ float shist[CELLS_PER_STRIP * NBINS];  // 224
    __shared__ float sgauss[CS * CS];                 // 64

    const int tid = threadIdx.x;
    const int b  = blockIdx.x / HC;      // image
    const int hc = blockIdx.x % HC;      // cell row
    const int r0 = hc * CS - 1;          // first halo row (image coords)

    // init histogram + gaussian weights
    if (tid < CELLS_PER_STRIP * NBINS) shist[tid] = 0.0f;
    if (tid < CS * CS) sgauss[tid] = gauss[tid];

    // stage gray tile: 10 rows x 226 cols, zero padding outside the image
    const size_t img_base = (size_t)b * 3u * IMG * IMG;
    for (int i = tid; i < 10 * 226; i += 256) {
        const int lr = i / 226;          // 0..9
        const int lc = i % 226;          // 0..225
        const int r = r0 + lr;
        const int c = lc - 1;
        float g = 0.0f;
        if (r >= 0 && r < IMG && c >= 0 && c < IMG) {
            const size_t p = img_base + (size_t)r * IMG + c;
            const float R = x[p];
            const float G = x[p + (size_t)IMG * IMG];
            const float B = x[p + 2u * (size_t)IMG * IMG];
            g = 0.2989f * R + 0.587f * G + 0.114f * B;
        }
        sgray[lr * 228 + lc] = g;
    }
    __syncthreads();

    // 8 x 224 pixels per strip; 7 per thread
    for (int p = tid; p < CS * IMG; p += 256) {
        const int r = p / IMG;           // 0..7 (row inside cell == row in strip)
        const int c = p % IMG;
        const float* g0 = &sgray[r * 228 + c];   // top-left of the 3x3 window
        const float a00 = g0[0],   a01 = g0[1],   a02 = g0[2];
        const float a10 = g0[228],                a12 = g0[230];
        const float a20 = g0[456], a21 = g0[457], a22 = g0[458];

        // cross-correlation with Sobel kernels (matches F.conv2d / lax.conv)
        const float gx = (a02 - a00) + 2.0f * (a12 - a10) + (a22 - a20);
        const float gy = (a20 - a00) + 2.0f * (a21 - a01) + (a22 - a02);

        const float mag = sqrtf(gx * gx + gy * gy + 1e-6f);
        float ang = atan2f(gy, gx);
        if (ang < 0.0f) ang += TWO_PI_F;
        const int bin = ((int)(ang * INV_BINW)) & (NBINS - 1);

        const float w = sgauss[((r & 7) << 3) | (c & 7)];
        // LDS float atomic scatter-add -> ds_add_f32
        atomicAdd(&shist[((c >> 3) << 3) | bin], mag * w);
    }
    __syncthreads();

    // write 28 cells x 8 bins
    float* o = out + (size_t)b * HIST_PER_IMG + (size_t)hc * (CELLS_PER_STRIP * NBINS);
    for (int i = tid; i < CELLS_PER_STRIP * NBINS; i += 256)
        o[i] = shist[i];
}

// ---------------------------------------------------------------------------
// Kernel 2: per-image L2 normalization; sum-of-squares via V_WMMA_F32_16X16X4_F32
// A = squared histogram values, B = ones  =>  D columns hold per-row sums
// (permutation-invariant in K, so exact regardless of operand lane layout).
// grid.x = bs; block = 256 (8 waves, each reduces a strided set of 64-chunks)
//
// The chunk loop bounds are forced into SGPRs via readfirstlane so the backend
// emits a scalar-branched loop: EXEC stays all-1s at every WMMA (ISA 7.12
// requirement), with no per-iteration exec-mask bookkeeping.
// ---------------------------------------------------------------------------
__global__ __launch_bounds__(256) void hog_norm_kernel(float* __restrict__ out)
{
    __shared__ float spart[8];
    __shared__ float sscale;

    const int tid  = threadIdx.x;
    const int lane = tid & 31;
    // wave id as a provably wave-uniform (SGPR) value
    const int wv = __builtin_amdgcn_readfirstlane(tid >> 5);
    float* row = out + (size_t)blockIdx.x * HIST_PER_IMG;

    const v2f ones = {1.0f, 1.0f};
    v8f c = {0.f, 0.f, 0.f, 0.f, 0.f, 0.f, 0.f, 0.f};

    // 6272 = 98 chunks of 64; wave w takes chunks w, w+8, ...
    for (int chunk = wv; chunk < HIST_PER_IMG / 64; chunk += 8) {
        const int base = chunk * 64;
        const float h0 = row[base + lane];
        const float h1 = row[base + 32 + lane];
        v2f a;
        a.x = h0 * h0;
        a.y = h1 * h1;
        // D = A * B + C  (accumulate row-sums of squares into C)
        c = __builtin_amdgcn_wmma_f32_16x16x4_f32(
                /*neg_a=*/false, a, /*neg_b=*/false, ones,
                /*c_mod=*/(short)0, c, /*reuse_a=*/false, /*reuse_b=*/false);
    }

    // Column N=0 of D lives in lane 0 (M=0..7 in c[0..7]) and lane 16 (M=8..15).
    float d = 0.0f;
    if (lane == 0 || lane == 16) {
        d = c[0] + c[1] + c[2] + c[3] + c[4] + c[5] + c[6] + c[7];
    }
    for (int off = 16; off > 0; off >>= 1)
        d += __shfl_down(d, off, 32);

    if (lane == 0) spart[wv] = d;
    __syncthreads();

    if (tid == 0) {
        float t = 0.0f;
        #pragma unroll
        for (int i = 0; i < 8; ++i) t += spart[i];
        sscale = 1.0f / (sqrtf(t) + 1e-6f);
    }
    __syncthreads();

    const float s = sscale;
    for (int i = tid; i < HIST_PER_IMG; i += 256)
        row[i] *= s;
}

// ---------------------------------------------------------------------------
extern "C" void kernel_launch(void* const* d_in, const int* in_sizes, int n_in,
                              void* d_out, int out_size, void* d_ws, size_t ws_size,
                              hipStream_t stream)
{
    const float* x     = (const float*)d_in[0];
    const float* gauss = (const float*)d_in[1];
    // d_in[2]/d_in[3] are the fixed Sobel kernels; baked into the kernel.
    float* out = (float*)d_out;

    const int bs = in_sizes[0] / (3 * IMG * IMG);

    hog_strip_kernel<<<dim3(bs * HC), dim3(256), 0, stream>>>(x, gauss, out);
    hog_norm_kernel<<<dim3(bs), dim3(256), 0, stream>>>(out);
}